// EdgeIndex_89352499626116
// MI455X (gfx1250) — compile-verified
//
#include <hip/hip_runtime.h>
#include <stdint.h>

#define N_PTS 8192
#define MAX_E (64 * N_PTS)      // 524288 padded edges
#define CUT2 25.0f              // cutoff^2
#define NCHUNK 8                // column chunks per row-block
#define TILES_PER_CHUNK 64      // 512 column tiles / 8 chunks

typedef __attribute__((ext_vector_type(2))) float v2f;
typedef __attribute__((ext_vector_type(8))) float v8f;

// ---------------------------------------------------------------- fill -1
__global__ void fill_neg1(int* __restrict__ out, int n) {
    int i = blockIdx.x * blockDim.x + threadIdx.x;
    if (i < n) out[i] = -1;
}

// ---------------------------------------------------------------- |p|^2
__global__ void sq_kernel(const float* __restrict__ pos, float* __restrict__ sq) {
    int i = blockIdx.x * blockDim.x + threadIdx.x;
    if (i < N_PTS) {
        float x = pos[3 * i + 0];
        float y = pos[3 * i + 1];
        float z = pos[3 * i + 2];
        sq[i] = x * x + y * y + z * z;
    }
}

// ---------------------------------------------------------------- count pass
// One wave per (row-block R of 16 rows, column chunk of 64 16-wide tiles).
// Per tile: one v_wmma_f32_16x16x4_f32 computes the 16x16 Gram block.
__global__ __launch_bounds__(32)
void count_kernel(const float* __restrict__ pos, const float* __restrict__ sq,
                  unsigned* __restrict__ cnt2) {
    const int lane = threadIdx.x;      // 0..31 (wave32)
    const int l    = lane & 15;
    const int half = lane >> 4;
    const int R     = blockIdx.y;      // row tile index (0..511)
    const int chunk = blockIdx.x;      // column chunk (0..7)

    // A operand: 16x4 f32. lanes 0-15 hold K=0 (v0), K=1 (v1) for M=l;
    // lanes 16-31 hold K=2 (v0), K=3 (v1). K=3 is the zero pad.
    const int iA = R * 16 + l;
    const float* pA = pos + 3 * iA;
    v2f a;
    a.x = half ? pA[2] : pA[0];
    a.y = half ? 0.0f : pA[1];

    // |p_i|^2 for the 8 rows this lane's accumulators correspond to
    const int imBase = R * 16 + 8 * half;
    float sqi[8];
#pragma unroll
    for (int v = 0; v < 8; ++v) sqi[v] = sq[imBase + v];

    unsigned rowcnt = 0;  // lanes 0..15: neighbor count of row R*16+lane in this chunk

    for (int t = 0; t < TILES_PER_CHUNK; ++t) {
        const int Ct = chunk * TILES_PER_CHUNK + t;
        const int j  = Ct * 16 + l;
        const float* pB = pos + 3 * j;
        // B operand: 4x16 f32. VGPR0: K=0 (lanes 0-15) / K=2 (lanes 16-31);
        // VGPR1: K=1 / K=3(zero pad). Column N = l.
        v2f b;
        b.x = half ? pB[2] : pB[0];
        b.y = half ? 0.0f : pB[1];
        const float sqj = sq[j];

        v8f c = {};
        c = __builtin_amdgcn_wmma_f32_16x16x4_f32(
                false, a, false, b, (short)0, c, false, false);

        // C layout: VGPR v -> row v (lanes 0-15), row v+8 (lanes 16-31), col = l
#pragma unroll
        for (int v = 0; v < 8; ++v) {
            const int   im   = imBase + v;
            const float d2   = sqi[v] + sqj - 2.0f * c[v];
            const bool  pred = (d2 < CUT2) && (im != j);
            const unsigned bal = (unsigned)__ballot(pred);
            const unsigned lo  = bal & 0xffffu;   // row v
            const unsigned hi  = bal >> 16;       // row v+8
            if (lane == v)     rowcnt += __popc(lo);
            if (lane == v + 8) rowcnt += __popc(hi);
        }
    }
    if (lane < 16) cnt2[(R * 16 + lane) * NCHUNK + chunk] = rowcnt;
}

// ---------------------------------------------------------------- exclusive scan
// Flattened (row, chunk) order == jnp.nonzero row-major emission order.
__global__ __launch_bounds__(256)
void scan_kernel(const unsigned* __restrict__ cnt2, unsigned* __restrict__ off2) {
    __shared__ unsigned seg[256];
    const int t   = threadIdx.x;
    const int n   = N_PTS * NCHUNK;   // 65536
    const int per = n / 256;          // 256
    unsigned s = 0;
    for (int k = 0; k < per; ++k) s += cnt2[t * per + k];
    seg[t] = s;
    __syncthreads();
    if (t == 0) {
        unsigned run = 0;
        for (int k = 0; k < 256; ++k) { unsigned tmp = seg[k]; seg[k] = run; run += tmp; }
    }
    __syncthreads();
    unsigned run = seg[t];
    for (int k = 0; k < per; ++k) { off2[t * per + k] = run; run += cnt2[t * per + k]; }
}

// ---------------------------------------------------------------- emit pass
// Same WMMA tiling; identical FP path => identical predicates as count pass.
// Per-row cursors live in lanes 0-15, broadcast with __shfl; each set bit is
// written at cursor + rank-of-bit (parallel, deterministic, lexicographic).
__global__ __launch_bounds__(32)
void emit_kernel(const float* __restrict__ pos, const float* __restrict__ sq,
                 const unsigned* __restrict__ off2,
                 int* __restrict__ out_rows, int* __restrict__ out_cols) {
    const int lane = threadIdx.x;
    const int l    = lane & 15;
    const int half = lane >> 4;
    const int R     = blockIdx.y;
    const int chunk = blockIdx.x;

    const int iA = R * 16 + l;
    const float* pA = pos + 3 * iA;
    v2f a;
    a.x = half ? pA[2] : pA[0];
    a.y = half ? 0.0f : pA[1];

    const int imBase = R * 16 + 8 * half;
    float sqi[8];
#pragma unroll
    for (int v = 0; v < 8; ++v) sqi[v] = sq[imBase + v];

    int cur = 0;
    if (lane < 16) cur = (int)off2[(R * 16 + lane) * NCHUNK + chunk];

    for (int t = 0; t < TILES_PER_CHUNK; ++t) {
        const int Ct = chunk * TILES_PER_CHUNK + t;
        const int j  = Ct * 16 + l;
        const float* pB = pos + 3 * j;
        v2f b;
        b.x = half ? pB[2] : pB[0];
        b.y = half ? 0.0f : pB[1];
        const float sqj = sq[j];

        v8f c = {};
        c = __builtin_amdgcn_wmma_f32_16x16x4_f32(
                false, a, false, b, (short)0, c, false, false);

#pragma unroll
        for (int v = 0; v < 8; ++v) {
            const int   im   = imBase + v;
            const float d2   = sqi[v] + sqj - 2.0f * c[v];
            const bool  pred = (d2 < CUT2) && (im != j);
            const unsigned bal = (unsigned)__ballot(pred);  // uniform across wave
            const unsigned lo  = bal & 0xffffu;
            const unsigned hi  = bal >> 16;

            const unsigned mymask = half ? hi : lo;
            const int base = __shfl(cur, v + 8 * half);     // cursor of my row
            if (mymask & (1u << l)) {
                const int p = base + __popc(mymask & ((1u << l) - 1u));
                if (p < MAX_E) {
                    out_rows[p] = im;
                    out_cols[p] = Ct * 16 + l;
                }
            }
            if (lane == v)     cur += __popc(lo);
            if (lane == v + 8) cur += __popc(hi);
        }
    }
}

// ---------------------------------------------------------------- launcher
extern "C" void kernel_launch(void* const* d_in, const int* in_sizes, int n_in,
                              void* d_out, int out_size, void* d_ws, size_t ws_size,
                              hipStream_t stream) {
    const float* pos = (const float*)d_in[0];
    int* out = (int*)d_out;     // int32 output: rows[MAX_E] then cols[MAX_E]

    char* ws = (char*)d_ws;
    float*    sq   = (float*)ws;                              //  32 KB
    unsigned* cnt2 = (unsigned*)(ws + 32 * 1024);             // 256 KB
    unsigned* off2 = (unsigned*)(ws + 32 * 1024 + 256 * 1024);// 256 KB

    fill_neg1<<<(2 * MAX_E + 255) / 256, 256, 0, stream>>>(out, 2 * MAX_E);
    sq_kernel<<<(N_PTS + 255) / 256, 256, 0, stream>>>(pos, sq);

    dim3 grid(NCHUNK, N_PTS / 16);   // (8, 512) -> 4096 waves per pass
    count_kernel<<<grid, 32, 0, stream>>>(pos, sq, cnt2);
    scan_kernel<<<1, 256, 0, stream>>>(cnt2, off2);
    emit_kernel<<<grid, 32, 0, stream>>>(pos, sq, off2, out, out + MAX_E);
}